// MMD_loss_26560077758605
// MI455X (gfx1250) — compile-verified
//
#include <hip/hip_runtime.h>
#include <hip/hip_bf16.h>

// ---------------- CDNA5 WMMA types ----------------
typedef float v2f __attribute__((ext_vector_type(2)));
typedef float v8f __attribute__((ext_vector_type(8)));
// GCC-style vector type matching the async-LDS builtin's parameter type
typedef int vi4 __attribute__((vector_size(4 * sizeof(int))));

#define N_ROWS   4096      // rows per matrix (source / target)
#define DIM      256       // feature dim
#define TOTAL    (2 * N_ROWS)
#define STRIDE   260       // padded LDS row stride (floats): 16B-aligned rows, conflict-free b64 reads
#define BTILE_FLOATS (32 * STRIDE)                   // one B buffer, in floats
#define JTILES_PER_BLOCK 16
#define GRID_X   (TOTAL / (32 * JTILES_PER_BLOCK))   // 16
#define GRID_Y   (TOTAL / 32)                        // 256
#define NPART    (GRID_X * GRID_Y)                   // 4096 partials

// --------- CDNA5 async global->LDS path (guarded; fallback = sync staging) ---------
#if defined(__gfx1250__) && \
    __has_builtin(__builtin_amdgcn_global_load_async_to_lds_b128) && \
    __has_builtin(__builtin_amdgcn_s_wait_asynccnt)
#define HAVE_ASYNC_LDS 1
#else
#define HAVE_ASYNC_LDS 0
#endif

__device__ __forceinline__ void copy16_to_lds(const float* __restrict__ g, float* l) {
#if HAVE_ASYNC_LDS
    // global_load_async_to_lds_b128: 16B/lane, tracked by ASYNCcnt, no VGPR data
    __builtin_amdgcn_global_load_async_to_lds_b128(
        (__attribute__((address_space(1))) vi4*)g,
        (__attribute__((address_space(3))) vi4*)l,
        /*imm offset=*/0, /*cpol=*/0);
#else
    *(float4*)l = *(const float4*)g;
#endif
}

__device__ __forceinline__ void wait_async_copies() {
#if HAVE_ASYNC_LDS
    __builtin_amdgcn_s_wait_asynccnt(0);
#endif
}

// Row i of the conceptual concat(source, target) matrix.
__device__ __forceinline__ const float* row_ptr(const float* __restrict__ s,
                                                const float* __restrict__ t, int i) {
    return (i < N_ROWS) ? (s + (size_t)i * DIM) : (t + (size_t)(i - N_ROWS) * DIM);
}

// ---------------- Stage 1: row squared-norms ----------------
__global__ void sq_kernel(const float* __restrict__ src, const float* __restrict__ tgt,
                          float* __restrict__ sq) {
    __shared__ float red[DIM];
    const int row = blockIdx.x;
    const float* p = row_ptr(src, tgt, row);
    float x = p[threadIdx.x];
    red[threadIdx.x] = x * x;
    __syncthreads();
    for (int s = DIM / 2; s > 0; s >>= 1) {
        if (threadIdx.x < (unsigned)s) red[threadIdx.x] += red[threadIdx.x + s];
        __syncthreads();
    }
    if (threadIdx.x == 0) sq[row] = red[0];
}

// ---------------- Stage 2: fused Gram + multi-sigma RBF + partial reduce ----------------
// Block = 128 threads (4 wave32). Block strip: 32 rows (I), staged once.
// Streams 16 J-tiles of 32 cols with async double-buffering.
// Each wave computes one 16x16 tile via v_wmma_f32_16x16x4_f32.
// NOTE: LDS pointers are plain variables derived straight from `smem` (no pointer
// arrays) so addrspace inference keeps the DS path, and fragment bases carry an
// 8B alignment assumption so v2f loads lower to (pairable) b64 DS ops.
__global__ void __launch_bounds__(128)
mmd_main(const float* __restrict__ src, const float* __restrict__ tgt,
         const float* __restrict__ sq, float* __restrict__ partials) {
    extern __shared__ float smem[];   // As[32*STRIDE] | Bbuf0[32*STRIDE] | Bbuf1[32*STRIDE]

    const int tid    = threadIdx.x;
    const int lane   = tid & 31;
    const int wave   = tid >> 5;             // 0..3
    const int lane15 = lane & 15;
    const int hi     = lane >> 4;            // 0: K={0,1}/M=0..7 ; 1: K={2,3}/M=8..15
    const int subI   = wave >> 1;            // 0..1
    const int subJ   = wave & 1;             // 0..1

    const int i0 = blockIdx.y * 32;          // strip base row
    const int jBase = blockIdx.x * JTILES_PER_BLOCK * 32;

    // ---- prologue: async-stage A strip + B tile 0 ----
    // 32 rows * 256 floats = 2048 float4; 128 threads -> 16 xfers each per tile
    #pragma unroll
    for (int it = 0; it < 16; ++it) {
        int idx  = tid + it * 128;
        int row  = idx >> 6;                 // /64 float4 per row
        int col4 = idx & 63;
        copy16_to_lds(row_ptr(src, tgt, i0 + row) + col4 * 4,
                      smem + row * STRIDE + col4 * 4);
    }
    #pragma unroll
    for (int it = 0; it < 16; ++it) {
        int idx  = tid + it * 128;
        int row  = idx >> 6;
        int col4 = idx & 63;
        copy16_to_lds(row_ptr(src, tgt, jBase + row) + col4 * 4,
                      smem + BTILE_FLOATS + row * STRIDE + col4 * 4);
    }

    // Hoist this wave's 8 row-norms (fixed across the J loop)
    float sqr[8];
    #pragma unroll
    for (int e = 0; e < 8; ++e) sqr[e] = sq[i0 + subI * 16 + e + 8 * hi];

    // A fragment base: 8B-aligned (row base is a multiple of 1040B, +hi*8B)
    const float* aRow = (const float*)__builtin_assume_aligned(
        smem + (subI * 16 + lane15) * STRIDE + 2 * hi, 8);
    const int bRowOff = (subJ * 16 + lane15) * STRIDE + 2 * hi;   // within a B buffer

    float blockSum = 0.0f;

    for (int jt = 0; jt < JTILES_PER_BLOCK; ++jt) {
        // integer-offset double-buffer select (keeps LDS addrspace inference intact)
        const int curOff = BTILE_FLOATS + (jt & 1) * BTILE_FLOATS;          // this tile's buffer
        const int nxtOff = curOff == BTILE_FLOATS ? 2 * BTILE_FLOATS : BTILE_FLOATS;
        const int j0  = jBase + jt * 32;

        // my async copies done -> barrier makes everyone's LDS writes visible,
        // and guarantees the buffer we are about to prefetch into is no longer read
        wait_async_copies();
        __syncthreads();

        // prefetch next B tile into the other buffer while we compute this one
        if (jt + 1 < JTILES_PER_BLOCK) {
            const int jn = j0 + 32;
            #pragma unroll
            for (int it = 0; it < 16; ++it) {
                int idx  = tid + it * 128;
                int row  = idx >> 6;
                int col4 = idx & 63;
                copy16_to_lds(row_ptr(src, tgt, jn + row) + col4 * 4,
                              smem + nxtOff + row * STRIDE + col4 * 4);
            }
        }

        const float* bRow = (const float*)__builtin_assume_aligned(
            smem + curOff + bRowOff, 8);

        // ---- fp32 WMMA Gram: G = A_rows . B_rows^T, K = 256 in steps of 4 ----
        v8f acc = {};
        #pragma unroll 8
        for (int k = 0; k < DIM; k += 4) {
            v2f a = *(const v2f*)(aRow + k);
            v2f b = *(const v2f*)(bRow + k);
            acc = __builtin_amdgcn_wmma_f32_16x16x4_f32(
                /*neg_a=*/false, a, /*neg_b=*/false, b,
                /*c_mod=*/(short)0, acc, /*reuse_a=*/false, /*reuse_b=*/false);
        }

        // ---- distances -> sum of 5 Gaussian kernels (one exp per element) ----
        // sigma = 1,2,4,8,16 ; t = exp(-l2/16) ; sum = t + t^2 + t^4 + t^8 + t^16
        const float sgn = ((i0 < N_ROWS) == (j0 < N_ROWS)) ? 1.0f : -1.0f;
        const float sqc = sq[j0 + subJ * 16 + lane15];
        float lsum = 0.0f;
        #pragma unroll
        for (int e = 0; e < 8; ++e) {
            float l2 = sqr[e] + sqc - 2.0f * acc[e];
            l2 = fmaxf(l2, 0.0f);
            float t1  = __expf(l2 * (-1.0f / 16.0f));
            float t2  = t1 * t1;
            float t4  = t2 * t2;
            float t8  = t4 * t4;
            float t16 = t8 * t8;
            lsum += t1 + t2 + t4 + t8 + t16;
        }
        blockSum += sgn * lsum;
    }

    // ---- deterministic per-block reduction (reuse LDS) ----
    __syncthreads();
    smem[tid] = blockSum;
    __syncthreads();
    for (int s = 64; s > 0; s >>= 1) {
        if (tid < s) smem[tid] += smem[tid + s];
        __syncthreads();
    }
    if (tid == 0) partials[blockIdx.y * gridDim.x + blockIdx.x] = smem[0];
}

// ---------------- Stage 3: fixed-order final reduce ----------------
__global__ void finalize_kernel(const float* __restrict__ partials, int nPart,
                                float* __restrict__ out) {
    __shared__ float red[256];
    float s = 0.0f;
    for (int i = threadIdx.x; i < nPart; i += 256) s += partials[i];
    red[threadIdx.x] = s;
    __syncthreads();
    for (int st = 128; st > 0; st >>= 1) {
        if (threadIdx.x < (unsigned)st) red[threadIdx.x] += red[threadIdx.x + st];
        __syncthreads();
    }
    if (threadIdx.x == 0)
        out[0] = red[0] * (1.0f / ((float)N_ROWS * (float)N_ROWS));
}

extern "C" void kernel_launch(void* const* d_in, const int* in_sizes, int n_in,
                              void* d_out, int out_size, void* d_ws, size_t ws_size,
                              hipStream_t stream) {
    (void)in_sizes; (void)n_in; (void)out_size; (void)ws_size;
    const float* src = (const float*)d_in[0];
    const float* tgt = (const float*)d_in[1];
    float* out = (float*)d_out;

    float* wsf      = (float*)d_ws;
    float* sq       = wsf;               // 8192 floats
    float* partials = wsf + TOTAL;       // 4096 floats

    // Stage 1: row norms
    sq_kernel<<<TOTAL, DIM, 0, stream>>>(src, tgt, sq);

    // Stage 2: fused WMMA MMD tiles (async double-buffered B)
    dim3 grid(GRID_X, GRID_Y);
    size_t lds = (size_t)3 * BTILE_FLOATS * sizeof(float);   // 99,840 B (< 320 KB WGP LDS)
    mmd_main<<<grid, 128, lds, stream>>>(src, tgt, sq, partials);

    // Stage 3: deterministic final reduction + mean scale
    finalize_kernel<<<1, 256, 0, stream>>>(partials, NPART, out);
}